// MLP_NL_82420422410796
// MI455X (gfx1250) — compile-verified
//
#include <hip/hip_runtime.h>
#include <hip/hip_bf16.h>
#include <math.h>

#define B_SZ   2048
#define N_SZ   8192
#define H_SZ   10
#define BETA_C 0.1f
#define TPB    256
#define TILE   1024            // floats per tile
#define NTILES (N_SZ / TILE)   // 8
#define EPT    (TILE / TPB)    // 4 elements / thread, one b128 per stream

// ---------------- CDNA5 async global->LDS helpers ----------------
typedef int v4i __attribute__((ext_vector_type(4)));
typedef __attribute__((address_space(1))) v4i gv4i;   // global (prints as __device__)
typedef __attribute__((address_space(3))) v4i lv4i;   // LDS    (prints as __shared__)

__device__ __forceinline__ void async_load_b128(const float* g, float* l) {
#if __has_builtin(__builtin_amdgcn_global_load_async_to_lds_b128)
    __builtin_amdgcn_global_load_async_to_lds_b128(
        (gv4i*)g, (lv4i*)l, /*offset=*/0, /*cpol=*/0);
#else
    unsigned lds_addr = (unsigned)(unsigned long long)l;
    asm volatile("global_load_async_to_lds_b128 %0, %1, off"
                 :: "v"(lds_addr), "v"(g) : "memory");
#endif
}

template <int C>
__device__ __forceinline__ void wait_async() {
#if __has_builtin(__builtin_amdgcn_s_wait_asynccnt)
    __builtin_amdgcn_s_wait_asynccnt(C);
#else
    asm volatile("s_wait_asynccnt %0" :: "n"(C) : "memory");
#endif
}

// ---------------- kernel: one block per batch row ----------------
__global__ __launch_bounds__(TPB) void frac_mlp_kernel(
    const float* __restrict__ y_plus,  // [B]
    const int*   __restrict__ idx,     // [B]
    const float* __restrict__ du_all,  // [B,N]
    const float* __restrict__ y_all,   // [B,N]
    const float* __restrict__ aw1, const float* __restrict__ ab1,
    const float* __restrict__ aw2, const float* __restrict__ ab2,
    const float* __restrict__ w1,  const float* __restrict__ b1,
    const float* __restrict__ w2,  const float* __restrict__ b2,
    float* __restrict__ out)           // [B]
{
    __shared__ __align__(16) float sh_y [2][TILE];
    __shared__ __align__(16) float sh_du[2][TILE];
    __shared__ float red[TPB / 32];

    const int b = blockIdx.x;
    const int t = threadIdx.x;
    const float* __restrict__ yrow  = y_all  + (size_t)b * N_SZ;
    const float* __restrict__ durow = du_all + (size_t)b * N_SZ;

    // ---- alpha = 1 - sigmoid(mlp(y_plus)) * (1 - exp(-beta*y_plus)) ----
    // Block-uniform: compiler scalarizes this to the SALU pipe.
    const float yp = y_plus[b];
    const float g  = 1.0f - __expf(-BETA_C * yp);
    float mm = ab2[0];
#pragma unroll
    for (int j = 0; j < H_SZ; ++j) {
        float h = fmaf(yp, aw1[j], ab1[j]);
        mm = fmaf(fmaxf(h, 0.0f), aw2[j], mm);
    }
    const float a    = 1.0f - (1.0f / (1.0f + expf(-mm))) * g;
    const float texp = 1.0f - a;
    const int   n    = idx[b];
    const float yn   = yrow[n];

    // ---- a == 1 (g == 0 exactly): d = du[n], skip the whole stream ----
    if (a == 1.0f) {
        if (t == 0) {
            const float d = durow[n];
            float acc = b2[0];
#pragma unroll
            for (int j = 0; j < H_SZ; ++j)
                acc = fmaf(fmaxf(fmaf(d, w1[j], b1[j]), 0.0f), w2[j], acc);
            out[b] = acc;
        }
        return;
    }
    const bool trap_mode = (a == 0.0f);   // block-uniform branch in hot loop

    // ---- prologue: stage tile 0 (2 async b128 per thread => 2/wave) ----
    async_load_b128(yrow  + t * EPT, &sh_y [0][t * EPT]);
    async_load_b128(durow + t * EPT, &sh_du[0][t * EPT]);

    // Telescoped main sum (merges ref's sum_main + last term):
    //   sum_{k=0}^{n-1} (yn - y[k])^texp * (du[k] - du[k-1]),  du[-1] := 0
    float acc = 0.0f;
    int cur = 0;

    for (int it = 0; it < NTILES; ++it) {
        const int gbase = it * TILE;
        if (it + 1 < NTILES) {
            const int nb = gbase + TILE;
            async_load_b128(yrow  + nb + t * EPT, &sh_y [cur ^ 1][t * EPT]);
            async_load_b128(durow + nb + t * EPT, &sh_du[cur ^ 1][t * EPT]);
            wait_async<2>();   // newest 2 in flight; current tile has landed
        } else {
            wait_async<0>();
        }
        __syncthreads();       // all 8 waves' async data visible in LDS

#pragma unroll
        for (int j = 0; j < EPT; ++j) {
            const int l = j * TPB + t;     // stride-TPB: bank-conflict-free
            const int k = gbase + l;
            if (k <= n - 1) {              // active region, base yn-y[k] > 0
                const float yk  = sh_y [cur][l];
                const float duk = sh_du[cur][l];
                if (trap_mode) {           // a == 0: cumtrapz only, no pow
                    float yk1, duk1;
                    if (l + 1 < TILE) { yk1 = sh_y[cur][l + 1]; duk1 = sh_du[cur][l + 1]; }
                    else              { yk1 = yrow[k + 1];      duk1 = durow[k + 1]; } // k+1 <= n
                    acc += (yk1 - yk) * (duk1 + duk) * 0.5f;
                } else {                   // common path: 1 log + 1 exp per element
                    const float dukm1 = (l > 0) ? sh_du[cur][l - 1]
                                                : (k > 0 ? durow[k - 1] : 0.0f);
                    acc += __expf(texp * __logf(yn - yk)) * (duk - dukm1);
                }
            }
        }
        __syncthreads();       // reads done before next prefetch overwrites
        cur ^= 1;
    }

    // ---- wave32 reduction, then cross-wave via LDS ----
#pragma unroll
    for (int off = 16; off > 0; off >>= 1)
        acc += __shfl_down(acc, off, 32);
    if ((t & 31) == 0) red[t >> 5] = acc;
    __syncthreads();

    if (t == 0) {
        float s = 0.0f;
#pragma unroll
        for (int w = 0; w < TPB / 32; ++w) s += red[w];

        // frac = exp(-lgamma(2-a)) * s ; n==0 gives s==0 -> frac==0 (matches ref)
        const float d = trap_mode ? s : expf(-lgammaf(2.0f - a)) * s;

        float acc2 = b2[0];
#pragma unroll
        for (int j = 0; j < H_SZ; ++j)
            acc2 = fmaf(fmaxf(fmaf(d, w1[j], b1[j]), 0.0f), w2[j], acc2);
        out[b] = acc2;
    }
}

extern "C" void kernel_launch(void* const* d_in, const int* in_sizes, int n_in,
                              void* d_out, int out_size, void* d_ws, size_t ws_size,
                              hipStream_t stream) {
    const float* y_plus = (const float*)d_in[0];
    const int*   i_idx  = (const int*)  d_in[1];
    const float* DU_DY  = (const float*)d_in[2];
    const float* Y      = (const float*)d_in[3];
    const float* aw1    = (const float*)d_in[4];
    const float* ab1    = (const float*)d_in[5];
    const float* aw2    = (const float*)d_in[6];
    const float* ab2    = (const float*)d_in[7];
    const float* w1     = (const float*)d_in[8];
    const float* b1     = (const float*)d_in[9];
    const float* w2     = (const float*)d_in[10];
    const float* b2     = (const float*)d_in[11];
    float* out = (float*)d_out;

    const int B = in_sizes[0];   // 2048 rows; one block per row
    frac_mlp_kernel<<<B, TPB, 0, stream>>>(y_plus, i_idx, DU_DY, Y,
                                           aw1, ab1, aw2, ab2,
                                           w1, b1, w2, b2, out);
}